// InfoNCE_13331578487378
// MI455X (gfx1250) — compile-verified
//
#include <hip/hip_runtime.h>
#include <math.h>

// InfoNCE (grouped) for MI455X / gfx1250. Bandwidth-bound: nk is 1.07 GB,
// AI ~ 0.5 FLOP/B -> read nk exactly once (async-to-LDS / non-temporal),
// fuse normalization, reduce on-chip. Matrix ops: V_WMMA_F32_16X16X4_F32.

#define N_TOTAL 8192
#define M_NEG   128
#define D_DIM   256
#define INV_T   10.0f   // 1 / TEMPERATURE

typedef __attribute__((ext_vector_type(2))) float v2f;
typedef __attribute__((ext_vector_type(4))) float f4;
typedef __attribute__((ext_vector_type(8))) float v8f;
typedef __attribute__((ext_vector_type(4))) int   v4i;

#if defined(__AMDGCN__) && __has_builtin(__builtin_amdgcn_global_load_async_to_lds_b128)
#define HAVE_ASYNC_LDS 1
// pointer to v4i in global (AS1) / LDS (AS3), per the builtin's prototype
typedef __attribute__((address_space(1))) v4i* gptr_v4i;
typedef __attribute__((address_space(3))) v4i* lptr_v4i;
#else
#define HAVE_ASYNC_LDS 0
#endif

template <int CNT>
__device__ __forceinline__ void wait_async_le() {
#if HAVE_ASYNC_LDS
#if __has_builtin(__builtin_amdgcn_s_wait_asynccnt)
    __builtin_amdgcn_s_wait_asynccnt((unsigned short)CNT);
#else
    asm volatile("s_wait_asynccnt %0" ::"i"(CNT) : "memory");
#endif
#endif
}

// Stage a 16-row x 64-col fp32 tile (row stride D_DIM) into wave-private LDS.
// Async path: 8 x global_load_async_to_lds_b128 (ASYNCcnt, no VGPR round-trip).
// Fallback:   8 NT b128 loads into regs, then 8 ds_store_b128 (keeps MLP).
__device__ __forceinline__ void stage_tile(const float* __restrict__ src,
                                           float* __restrict__ dstLds, int lane) {
#if HAVE_ASYNC_LDS
#pragma unroll
    for (int i = 0; i < 8; ++i) {
        int idx = lane + i * 32;          // float4 index 0..255
        int r   = idx >> 4;               // row 0..15
        int c4  = idx & 15;               // float4 within row
        const float* g = src + (long)r * D_DIM + c4 * 4;
        float*       l = dstLds + r * 64 + c4 * 4;
        __builtin_amdgcn_global_load_async_to_lds_b128(
            (gptr_v4i)(__SIZE_TYPE__)g,
            (lptr_v4i)(unsigned)(__SIZE_TYPE__)l, 0, 0);
    }
#else
    f4 tmp[8];
#pragma unroll
    for (int i = 0; i < 8; ++i) {
        int idx = lane + i * 32;
        int r   = idx >> 4;
        int c4  = idx & 15;
        tmp[i]  = __builtin_nontemporal_load((const f4*)(src + (long)r * D_DIM + c4 * 4));
    }
#pragma unroll
    for (int i = 0; i < 8; ++i) {
        int idx = lane + i * 32;
        int r   = idx >> 4;
        int c4  = idx & 15;
        *(f4*)(dstLds + r * 64 + c4 * 4) = tmp[i];
    }
#endif
}

// ---- workspace layout (floats) ----
//   [0,128)          denom[m]
//   [128,256)        pos_grouped[m]
//   [256,256+8192)   qinv[n]

__global__ void zero_ws_kernel(float* ws) {
    ws[threadIdx.x] = 0.0f;           // 256 threads: denom + pos_grouped
}

__global__ __launch_bounds__(256) void qinv_kernel(const float* __restrict__ q,
                                                   float* __restrict__ qinv) {
    const int wave = threadIdx.x >> 5;
    const int lane = threadIdx.x & 31;
    const int n = blockIdx.x * 8 + wave;
    if (n >= N_TOTAL) return;
    const float* row = q + (long)n * D_DIM;
    float ss = 0.0f;
#pragma unroll
    for (int k = 0; k < D_DIM / 32; ++k) {
        float v = row[lane + k * 32];
        ss += v * v;
    }
#pragma unroll
    for (int off = 16; off >= 1; off >>= 1) ss += __shfl_xor(ss, off, 32);
    if (lane == 0) qinv[n] = 1.0f / fmaxf(sqrtf(ss), 1e-12f);
}

// Dominant kernel: denom[m] = sum_n exp( (q_n . nk_nm) / (|q_n||nk_nm|) * 10 )
// 8 waves/block, wave w owns m-tile [16w,16w+16); blocks stride over n.
// Per n: double-buffered 16x64 nk chunks (async copy ahead of WMMA compute).
__global__ __launch_bounds__(256) void neg_denom_kernel(const float* __restrict__ q,
                                                        const float* __restrict__ nk,
                                                        const float* __restrict__ qinv,
                                                        float* __restrict__ denom) {
    __shared__ __align__(16) float qs[D_DIM];               // normalized q_n
    __shared__ __align__(16) float tile[8][2][16 * 64];     // per-wave double buffer
    __shared__ __align__(16) float ctile[8][16 * 16];       // per-wave WMMA D staging

    const int tid  = threadIdx.x;
    const int wave = tid >> 5;
    const int lane = tid & 31;
    const int lrow = lane & 15;
    const bool hi  = lane >= 16;
    const int m0   = wave * 16;
    float* tw0 = tile[wave][0];
    float* tw1 = tile[wave][1];
    float* cw  = ctile[wave];

    float dsum = 0.0f;   // lanes 0..15: running denom partial for m = m0+lrow

    for (int n = blockIdx.x; n < N_TOTAL; n += gridDim.x) {
        __syncthreads();
        qs[tid] = q[(long)n * D_DIM + tid] * qinv[n];   // stage q_n / ||q_n||
        __syncthreads();

        const float* nkbase = nk + ((long)n * M_NEG + m0) * D_DIM;

        if (n + (int)gridDim.x < N_TOTAL)
            __builtin_prefetch(nk + ((long)(n + gridDim.x) * M_NEG + m0) * D_DIM, 0, 1);

        v8f c = {};        // 16x16 f32 accumulator (all columns identical)
        float ss = 0.0f;   // per-lane partial sum of nk^2 (half-rows)

        stage_tile(nkbase, tw0, lane);                  // chunk 0 -> buf 0
#pragma unroll
        for (int kc = 0; kc < 4; ++kc) {
            float* cur = (kc & 1) ? tw1 : tw0;
            float* nxt = (kc & 1) ? tw0 : tw1;
            if (kc < 3) stage_tile(nkbase + (kc + 1) * 64, nxt, lane);
            if (kc < 3) wait_async_le<8>();             // drain chunk kc only
            else        wait_async_le<0>();

#pragma unroll
            for (int s = 0; s < 16; ++s) {
                // A fragment: lanes 0-15 hold (k0,k0+1), lanes 16-31 hold (k0+2,k0+3)
                int kk = 4 * s + (hi ? 2 : 0);
                float a0 = cur[lrow * 64 + kk];
                float a1 = cur[lrow * 64 + kk + 1];
                ss += a0 * a0 + a1 * a1;
                // B fragment: broadcast q -> every output column is the logit
                int kq = kc * 64 + kk;
                float b0 = qs[kq];
                float b1 = qs[kq + 1];
                v2f a; a.x = a0; a.y = a1;
                v2f b; b.x = b0; b.y = b1;
                c = __builtin_amdgcn_wmma_f32_16x16x4_f32(
                        false, a, false, b, (short)0, c, false, false);
            }
        }

        // D layout: VGPR v = row (v + 8*hi), lanes 0-15 = columns. Stage, read col 0.
#pragma unroll
        for (int v = 0; v < 8; ++v)
            cw[(v + (hi ? 8 : 0)) * 16 + lrow] = c[v];

        ss += __shfl_xor(ss, 16, 32);   // full ||nk_row||^2 for row lrow

        float raw = cw[lrow * 16 + 0];                   // q_hat . nk_row
        float z   = raw * rsqrtf(fmaxf(ss, 1e-24f));     // / ||nk_row||
        float e   = __expf(z * INV_T);
        if (!hi) dsum += e;
    }

    if (lane < 16) atomicAdd(&denom[m0 + lrow], dsum);
}

// pos_grouped[label[n]] += exp( (q_n . pk_n) / (|q_n||pk_n|) * 10 )
__global__ __launch_bounds__(256) void pos_kernel(const float* __restrict__ q,
                                                  const float* __restrict__ pk,
                                                  const int* __restrict__ labels,
                                                  const float* __restrict__ qinv,
                                                  float* __restrict__ posg) {
    const int wave = threadIdx.x >> 5;
    const int lane = threadIdx.x & 31;
    const int n = blockIdx.x * 8 + wave;
    if (n >= N_TOTAL) return;
    const float* qr = q  + (long)n * D_DIM;
    const float* pr = pk + (long)n * D_DIM;
    float dp = 0.0f, pp = 0.0f;
#pragma unroll
    for (int k = 0; k < D_DIM / 32; ++k) {
        float qv = qr[lane + k * 32];
        float pv = pr[lane + k * 32];
        dp += qv * pv;
        pp += pv * pv;
    }
#pragma unroll
    for (int off = 16; off >= 1; off >>= 1) {
        dp += __shfl_xor(dp, off, 32);
        pp += __shfl_xor(pp, off, 32);
    }
    if (lane == 0) {
        float z = dp * qinv[n] / fmaxf(sqrtf(pp), 1e-12f);
        atomicAdd(&posg[labels[n]], __expf(z * INV_T));
    }
}

// out = -sum(log(where(pg!=0, pg/denom, 1))) / max(count(pg!=0), 1)
__global__ void finalize_kernel(const float* __restrict__ denom,
                                const float* __restrict__ posg,
                                float* __restrict__ out) {
    __shared__ float ssum[4];
    __shared__ int   scnt[4];
    const int t = threadIdx.x;            // 128 threads
    const int wave = t >> 5, lane = t & 31;
    float pg = posg[t];
    int   valid = (pg != 0.0f) ? 1 : 0;
    float term  = valid ? logf(pg / denom[t]) : 0.0f;
#pragma unroll
    for (int off = 16; off >= 1; off >>= 1) {
        term  += __shfl_xor(term, off, 32);
        valid += __shfl_xor(valid, off, 32);
    }
    if (lane == 0) { ssum[wave] = term; scnt[wave] = valid; }
    __syncthreads();
    if (t == 0) {
        float s = ssum[0] + ssum[1] + ssum[2] + ssum[3];
        int  nv = scnt[0] + scnt[1] + scnt[2] + scnt[3];
        if (nv < 1) nv = 1;
        out[0] = -s / (float)nv;
    }
}

extern "C" void kernel_launch(void* const* d_in, const int* in_sizes, int n_in,
                              void* d_out, int out_size, void* d_ws, size_t ws_size,
                              hipStream_t stream) {
    const float* q      = (const float*)d_in[0];   // [8192, 256]
    const float* pk     = (const float*)d_in[1];   // [8192, 256]
    const int*   labels = (const int*)  d_in[2];   // [8192] (jax int64 -> int32)
    const float* nk     = (const float*)d_in[3];   // [8192, 128, 256]
    float* out = (float*)d_out;
    float* ws  = (float*)d_ws;

    float* denom = ws;          // 128
    float* posg  = ws + 128;    // 128
    float* qinv  = ws + 256;    // 8192

    zero_ws_kernel  <<<1, 256, 0, stream>>>(ws);
    qinv_kernel     <<<N_TOTAL / 8, 256, 0, stream>>>(q, qinv);
    neg_denom_kernel<<<1024, 256, 0, stream>>>(q, nk, qinv, denom);
    pos_kernel      <<<N_TOTAL / 8, 256, 0, stream>>>(q, pk, labels, qinv, posg);
    finalize_kernel <<<1, 128, 0, stream>>>(denom, posg, out);
}